// KNO2d_64707977282080
// MI455X (gfx1250) — compile-verified
//
#include <hip/hip_runtime.h>
#include <math.h>

#define BB 8
#define CIN 3
#define CHN 64
#define HH 256
#define WW 256
#define NKX 8
#define NLAYERS 6

typedef float v2f __attribute__((ext_vector_type(2)));
typedef float v8f __attribute__((ext_vector_type(8)));

#define TWO_PI 6.28318530717958647692f

// ---------------------------------------------------------------------------
// Kernel 1: encoder (tanh(conv1x1 3->64)) + fused w-direction DFT projection.
// One workgroup per (b,h) row. h0 is NOT materialized (recomputed in decoder);
// only the 4MB mode-projection partials Aw[b][h][c][ky] are written.
// ---------------------------------------------------------------------------
__global__ __launch_bounds__(256) void k_encoder_project(
    const float* __restrict__ x, const float* __restrict__ enc_w,
    const float* __restrict__ enc_b, float2* __restrict__ Aw)
{
    __shared__ float hlds[CHN * 257];   // padded rows -> conflict-free column reads
    __shared__ float tab[WW * 8];       // cos/sin for ky=0..3
    __shared__ float ew[CHN * 3];
    __shared__ float eb[CHN];

    const int wg = blockIdx.x;          // b*H + h
    const int b = wg / HH, h = wg % HH;
    const int t = threadIdx.x;          // 0..255

    if (t < CHN * 3) ew[t] = enc_w[t];
    if (t < CHN)     eb[t] = enc_b[t];
    {
        const int w = t;
        #pragma unroll
        for (int ky = 0; ky < 4; ++ky) {
            float s, c;
            __sincosf(TWO_PI * (float)(ky * w) / 256.0f, &s, &c);
            tab[w * 8 + 2 * ky]     = c;
            tab[w * 8 + 2 * ky + 1] = s;
        }
    }
    __syncthreads();

    // phase 1: encoder for one row, all 64 channels (into LDS only)
    {
        const int w = t;
        const size_t xbase = (size_t)b * CIN * HH * WW + (size_t)h * WW + w;
        const float x0 = x[xbase];
        const float x1 = x[xbase + (size_t)HH * WW];
        const float x2 = x[xbase + 2 * (size_t)HH * WW];
        for (int o = 0; o < CHN; ++o) {
            float v = fmaf(x2, ew[o*3+2], fmaf(x1, ew[o*3+1], fmaf(x0, ew[o*3+0], eb[o])));
            hlds[o * 257 + w] = tanhf(v);
        }
    }
    __syncthreads();

    // phase 2: Aw[h,ky] = sum_w h0[h,w] * e^{-2pi i ky w / W}
    {
        const int c = t >> 2, ky = t & 3;
        float re = 0.f, im = 0.f;
        for (int w = 0; w < WW; ++w) {
            const float v = hlds[c * 257 + w];
            re = fmaf(v,  tab[w*8 + 2*ky],     re);
            im = fmaf(-v, tab[w*8 + 2*ky + 1], im);
        }
        Aw[(((size_t)b * HH + h) * CHN + c) * 4 + ky] = make_float2(re, im);
    }
}

// ---------------------------------------------------------------------------
// Kernel 2: h-direction DFT: z0[b,c,kxi,ky] = sum_h Aw * e^{-2pi i kx h / H}
// kxi 0..3 -> kx 0..3 ; kxi 4..7 -> kx 252..255 (== kxi-8 mod 256)
// ---------------------------------------------------------------------------
__global__ __launch_bounds__(256) void k_project_kx(
    const float2* __restrict__ Aw, float2* __restrict__ z0)
{
    const int idx = blockIdx.x * 256 + threadIdx.x;   // B*CHN*8*4 = 16384
    const int ky  = idx & 3;
    const int kxi = (idx >> 2) & 7;
    const int c   = (idx >> 5) & 63;
    const int b   = idx >> 11;
    const int kxs = (kxi < 4) ? kxi : kxi - 8;
    float re = 0.f, im = 0.f;
    for (int h = 0; h < HH; ++h) {
        const float2 a = Aw[(((size_t)b * HH + h) * CHN + c) * 4 + ky];
        float s, cth;
        __sincosf(TWO_PI * (float)(kxs * h) / 256.0f, &s, &cth);
        // (ar + i ai) * e^{-i th}
        re = fmaf(a.x, cth, fmaf(a.y, s, re));
        im = fmaf(a.y, cth, fmaf(-a.x, s, im));
    }
    z0[(((size_t)b * CHN + c) * NKX + kxi) * 4 + ky] = make_float2(re, im);
}

// ---------------------------------------------------------------------------
// Kernel 3: six Koopman layers entirely in mode space.
//   z <- z + S(Mz),  wsum += Mz.   S = P∘R Hermitian fixup on ky=0 column:
//   kx=0: Re ; kx=252: *0.5 (partner kx=4 outside mode set) ;
//   (1,255),(2,254),(3,253): paired conjugate average.
// One workgroup per (b, group); 29 groups cover all 32 modes.
// ---------------------------------------------------------------------------
__global__ __launch_bounds__(256) void k_mode_iter(
    const float2* __restrict__ z0,
    const float* __restrict__ w1r, const float* __restrict__ w1i,
    const float* __restrict__ w2r, const float* __restrict__ w2i,
    float2* __restrict__ wsum)
{
    __shared__ float WAr[4096], WAi[4096], WBr[4096], WBi[4096];
    __shared__ float zAr[64], zAi[64], zBr[64], zBi[64];

    const int wg = blockIdx.x;
    const int b = wg / 29;
    const int g = wg % 29;
    int kxA, kxB = -1, ky, type;
    if (g < 24)      { ky = 1 + g / 8; kxA = g % 8;       type = 0; } // identity S
    else if (g == 24){ ky = 0; kxA = 0;                   type = 1; } // Re
    else if (g == 25){ ky = 0; kxA = 4;                   type = 2; } // *0.5 (kx=252)
    else             { ky = 0; kxA = g - 25; kxB = 8 - (g - 25); type = 3; } // pairs

    const int t = threadIdx.x;
    for (int idx = t; idx < 4096; idx += 256) {
        {
            const float* wr = (kxA < 4) ? w1r : w2r;
            const float* wi = (kxA < 4) ? w1i : w2i;
            const int xx = (kxA < 4) ? kxA : kxA - 4;
            const size_t off = (size_t)idx * 16 + xx * 4 + ky;
            WAr[idx] = wr[off]; WAi[idx] = wi[off];
        }
        if (type == 3) {
            const float* wr = (kxB < 4) ? w1r : w2r;
            const float* wi = (kxB < 4) ? w1i : w2i;
            const int xx = (kxB < 4) ? kxB : kxB - 4;
            const size_t off = (size_t)idx * 16 + xx * 4 + ky;
            WBr[idx] = wr[off]; WBi[idx] = wi[off];
        }
    }
    if (t < 64) {
        const float2 z = z0[(((size_t)b * CHN + t) * NKX + kxA) * 4 + ky];
        zAr[t] = z.x; zAi[t] = z.y;
        if (type == 3) {
            const float2 zb = z0[(((size_t)b * CHN + t) * NKX + kxB) * 4 + ky];
            zBr[t] = zb.x; zBi[t] = zb.y;
        }
    }
    __syncthreads();

    float wsAr = 0.f, wsAi = 0.f, wsBr = 0.f, wsBi = 0.f;
    for (int l = 0; l < NLAYERS; ++l) {
        float mAr = 0.f, mAi = 0.f, mBr = 0.f, mBi = 0.f;
        if (t < 64) {
            for (int i = 0; i < 64; ++i) {
                const float wr = WAr[i * 64 + t], wi = WAi[i * 64 + t];
                mAr = fmaf(zAr[i], wr, fmaf(-zAi[i], wi, mAr));
                mAi = fmaf(zAr[i], wi, fmaf( zAi[i], wr, mAi));
            }
            if (type == 3) {
                for (int i = 0; i < 64; ++i) {
                    const float wr = WBr[i * 64 + t], wi = WBi[i * 64 + t];
                    mBr = fmaf(zBr[i], wr, fmaf(-zBi[i], wi, mBr));
                    mBi = fmaf(zBr[i], wi, fmaf( zBi[i], wr, mBi));
                }
            }
        }
        __syncthreads();
        if (t < 64) {
            wsAr += mAr; wsAi += mAi;
            if (type == 0)      { zAr[t] += mAr;        zAi[t] += mAi; }
            else if (type == 1) { zAr[t] += mAr; }
            else if (type == 2) { zAr[t] += 0.5f * mAr; zAi[t] += 0.5f * mAi; }
            else {
                wsBr += mBr; wsBi += mBi;
                zAr[t] += 0.5f * (mAr + mBr); zAi[t] += 0.5f * (mAi - mBi);
                zBr[t] += 0.5f * (mBr + mAr); zBi[t] += 0.5f * (mBi - mAi);
            }
        }
        __syncthreads();
    }
    if (t < 64) {
        wsum[(((size_t)b * CHN + t) * NKX + kxA) * 4 + ky] = make_float2(wsAr, wsAi);
        if (type == 3)
            wsum[(((size_t)b * CHN + t) * NKX + kxB) * 4 + ky] = make_float2(wsBr, wsBi);
    }
}

// ---------------------------------------------------------------------------
// Kernel 4: partial reconstruction g[b,h,c,ky] = sum_kx wsum * e^{+2pi i kx h/H}
// ---------------------------------------------------------------------------
__global__ __launch_bounds__(256) void k_recon_g(
    const float2* __restrict__ wsum, float2* __restrict__ g)
{
    const int idx = blockIdx.x * 256 + threadIdx.x; // B*H*CHN*4 = 524288
    const int ky = idx & 3;
    const int c  = (idx >> 2) & 63;
    const int h  = (idx >> 8) & 255;
    const int b  = idx >> 16;
    float re = 0.f, im = 0.f;
    #pragma unroll
    for (int kxi = 0; kxi < 8; ++kxi) {
        const int kxs = (kxi < 4) ? kxi : kxi - 8;
        const float2 wv = wsum[(((size_t)b * CHN + c) * NKX + kxi) * 4 + ky];
        float s, cc;
        __sincosf(TWO_PI * (float)(kxs * h) / 256.0f, &s, &cc);
        re = fmaf(wv.x, cc, fmaf(-wv.y, s, re));
        im = fmaf(wv.x, s,  fmaf( wv.y, cc, im));
    }
    g[idx] = make_float2(re, im);
}

// ---------------------------------------------------------------------------
// Kernel 5: fused decoder. Recomputes the encoder from x (L2-resident, 6.3MB)
// instead of re-reading a 134MB h0 field:
//   t = 2*tanh(enc@x+b) + R(wsum)  ->  u = tanh(dec1 @ t)  ->  out = dec2 @ u
// dec1 (pixels x 64) @ (64 x 64) via V_WMMA_F32_16X16X4_F32, full f32.
// 128 pixels per workgroup, 8 waves; each wave owns one 16-pixel M-tile and
// all four 16-outch N-tiles -> A fragment reused across 4 WMMAs per K-step.
// ---------------------------------------------------------------------------
__global__ __launch_bounds__(256) void k_decoder(
    const float* __restrict__ x, const float2* __restrict__ g,
    const float* __restrict__ enc_w, const float* __restrict__ enc_b,
    const float* __restrict__ dec1_w, const float* __restrict__ dec1_b,
    const float* __restrict__ dec2_w, const float* __restrict__ dec2_b,
    float* __restrict__ out)
{
    __shared__ float tl[128 * 66];   // t field, [pix][c], padded stride 66
    __shared__ float ul[128 * 66];   // u field, [pix][o]
    __shared__ float wl[64 * 66];    // dec1_w, [o][c] == B[k][n] transposed view
    __shared__ float gl[512];        // g row (64 c * 4 ky * re/im)
    __shared__ float ew[CHN * 3];
    __shared__ float eb[CHN];
    __shared__ float d1b[64];
    __shared__ float d2w[64];

    const int wg = blockIdx.x;
    const int half = wg & 1;
    const int h = (wg >> 1) & 255;
    const int b = wg >> 9;
    const int w0 = half * 128;
    const int t = threadIdx.x;

    for (int idx = t; idx < 4096; idx += 256)
        wl[(idx >> 6) * 66 + (idx & 63)] = dec1_w[idx];
    if (t < CHN * 3) ew[t] = enc_w[t];
    if (t < 64) { eb[t] = enc_b[t]; d1b[t] = dec1_b[t]; d2w[t] = dec2_w[t]; }
    {
        const float* gf = (const float*)g;
        const size_t base = ((size_t)b * HH + h) * (CHN * 8);
        for (int idx = t; idx < 512; idx += 256) gl[idx] = gf[base + idx];
    }
    __syncthreads();

    // phase A: t[c][pix] = 2*h0 + (1/HW)*(Re g0 + 2*sum_ky Re(g_ky e^{i ky w th}))
    // with h0 recomputed from x on the fly.
    {
        const int pix = t & 127;
        const int coff = t >> 7;
        const int w = w0 + pix;
        const size_t xbase = (size_t)b * CIN * HH * WW + (size_t)h * WW + w;
        const float x0 = x[xbase];
        const float x1 = x[xbase + (size_t)HH * WW];
        const float x2 = x[xbase + 2 * (size_t)HH * WW];
        float c1, s1, c2, s2, c3, s3;
        __sincosf(TWO_PI * (float)w       / 256.0f, &s1, &c1);
        __sincosf(TWO_PI * (float)(2 * w) / 256.0f, &s2, &c2);
        __sincosf(TWO_PI * (float)(3 * w) / 256.0f, &s3, &c3);
        const float inv = 1.0f / 65536.0f;
        for (int cc = 0; cc < 32; ++cc) {
            const int c = cc * 2 + coff;
            const float y = gl[c*8 + 0]
                + 2.0f * ( gl[c*8+2]*c1 - gl[c*8+3]*s1
                         + gl[c*8+4]*c2 - gl[c*8+5]*s2
                         + gl[c*8+6]*c3 - gl[c*8+7]*s3 );
            float hv = fmaf(x2, ew[c*3+2], fmaf(x1, ew[c*3+1], fmaf(x0, ew[c*3+0], eb[c])));
            hv = tanhf(hv);
            tl[pix * 66 + c] = 2.0f * hv + inv * y;
        }
    }
    __syncthreads();

    // phase B: dec1 via WMMA f32 16x16x4.  A: lanes0-15 K{0,1}, lanes16-31 K{2,3}.
    const int lane  = t & 31;
    const int wave  = t >> 5;
    const int mrow  = lane & 15;
    const int khalf = lane >> 4;
    v8f acc[4];
    #pragma unroll
    for (int j = 0; j < 4; ++j) {
        const int tt = wave * 4 + j;
        const int mt = tt >> 2, nt = tt & 3;
        v8f c = {};
        #pragma unroll
        for (int kk = 0; kk < 16; ++kk) {
            const int kb = kk * 4 + khalf * 2;
            v2f a, bf;
            a.x  = tl[(mt * 16 + mrow) * 66 + kb];
            a.y  = tl[(mt * 16 + mrow) * 66 + kb + 1];
            bf.x = wl[(nt * 16 + mrow) * 66 + kb];
            bf.y = wl[(nt * 16 + mrow) * 66 + kb + 1];
            c = __builtin_amdgcn_wmma_f32_16x16x4_f32(
                    false, a, false, bf, (short)0, c, false, false);
        }
        acc[j] = c;
    }
    // bias + tanh, scatter u tiles to LDS (C layout: lanes0-15 M=r, lanes16-31 M=8+r)
    #pragma unroll
    for (int j = 0; j < 4; ++j) {
        const int tt = wave * 4 + j;
        const int mt = tt >> 2, nt = tt & 3;
        const int o = nt * 16 + mrow;
        const float bias = d1b[o];
        #pragma unroll
        for (int r = 0; r < 8; ++r) {
            const int pix = mt * 16 + khalf * 8 + r;
            ul[pix * 66 + o] = tanhf(acc[j][r] + bias);
        }
    }
    __syncthreads();

    // phase C: dec2 (64 -> 1)
    if (t < 128) {
        float s = dec2_b[0];
        for (int o = 0; o < 64; ++o) s = fmaf(d2w[o], ul[t * 66 + o], s);
        out[(size_t)b * (HH * WW) + (size_t)h * WW + w0 + t] = s;
    }
}

// ---------------------------------------------------------------------------
extern "C" void kernel_launch(void* const* d_in, const int* in_sizes, int n_in,
                              void* d_out, int out_size, void* d_ws, size_t ws_size,
                              hipStream_t stream) {
    const float* x      = (const float*)d_in[0];
    const float* enc_w  = (const float*)d_in[1];
    const float* enc_b  = (const float*)d_in[2];
    const float* dec1_w = (const float*)d_in[3];
    const float* dec1_b = (const float*)d_in[4];
    const float* dec2_w = (const float*)d_in[5];
    const float* dec2_b = (const float*)d_in[6];
    const float* w1r    = (const float*)d_in[7];
    const float* w1i    = (const float*)d_in[8];
    const float* w2r    = (const float*)d_in[9];
    const float* w2i    = (const float*)d_in[10];

    char* ws = (char*)d_ws;
    float2* Aw   = (float2*)(ws);               // 4,194,304 B
    float2* z0   = (float2*)(ws + 4194304);     //   131,072 B
    float2* wsum = (float2*)(ws + 4325376);     //   131,072 B
    float2* g    = (float2*)(ws + 4456448);     // 4,194,304 B
    float*  outp = (float*)d_out;

    hipLaunchKernelGGL(k_encoder_project, dim3(BB * HH), dim3(256), 0, stream,
                       x, enc_w, enc_b, Aw);
    hipLaunchKernelGGL(k_project_kx, dim3(64), dim3(256), 0, stream, Aw, z0);
    hipLaunchKernelGGL(k_mode_iter, dim3(BB * 29), dim3(256), 0, stream,
                       z0, w1r, w1i, w2r, w2i, wsum);
    hipLaunchKernelGGL(k_recon_g, dim3(2048), dim3(256), 0, stream, wsum, g);
    hipLaunchKernelGGL(k_decoder, dim3(BB * HH * 2), dim3(256), 0, stream,
                       x, g, enc_w, enc_b, dec1_w, dec1_b, dec2_w, dec2_b, outp);
}